// NeuralCDE_69664369541435
// MI455X (gfx1250) — compile-verified
//
#include <hip/hip_runtime.h>
#include <hip/hip_bf16.h>

// Neural CDE (RK4 over Hermite cubic path) for MI455X / gfx1250.
// f32 WMMA (V_WMMA_F32_16X16X4_F32) keeps reference precision; workload is
// latency-bound (2044 serial GEMM stages), so we minimize the dependent-WMMA
// critical path: K is split into independent accumulator chains per tile.

typedef __attribute__((ext_vector_type(2))) float v2f;
typedef __attribute__((ext_vector_type(8))) float v8f;

#define LSEQ 512
#define CCH  8
#define HH   32
#define MLPW 128
#define ONN  256   // C*H
#define MT   16    // batch rows per workgroup

__device__ __forceinline__ float tanh_fast(float x) {
  // tanh(x) = 1 - 2/(exp(2x)+1); saturates correctly at +-inf.
  float e = __expf(2.0f * x);
  return 1.0f - 2.0f * __builtin_amdgcn_rcpf(e + 1.0f);
}

// A fragment (16x4 f32): lane l -> row l&15; VGPR0/1 hold K = 2*(l>=16) + {0,1}.
__device__ __forceinline__ v2f frag_a(const float* base, int stride, int k0, int lane) {
  int m  = lane & 15;
  int kk = k0 + 2 * (lane >> 4);
  const float* p = base + m * stride + kk;
  v2f a; a.x = p[0]; a.y = p[1];
  return a;
}

// B fragment (4x16 f32, stored [K][N]): lane l -> col l&15; K = 2*(l>=16) + {0,1}.
__device__ __forceinline__ v2f frag_b(const float* base, int stride, int k0, int n0, int lane) {
  int n  = n0 + (lane & 15);
  int kk = k0 + 2 * (lane >> 4);
  const float* p = base + kk * stride + n;
  v2f b; b.x = p[0]; b.y = p[stride];
  return b;
}

__global__ __launch_bounds__(256, 1) void cde_rk4_kernel(
    const float* __restrict__ X, const float* __restrict__ Winit,
    const float* __restrict__ binit, const float* __restrict__ W1,
    const float* __restrict__ b1, const float* __restrict__ W2,
    const float* __restrict__ b2, float* __restrict__ out)
{
  // Odd row strides -> conflict-free column (A) reads; +1 pad on B panels
  // separates the two half-wave K rows into distinct banks.
  __shared__ float W2t[MLPW][ONN + 1];   // W2^T  [k][n]  (128.5 KB)
  __shared__ float W1t[HH][MLPW + 1];    // W1^T  [k][n]
  __shared__ float hS [MT][MLPW + 1];    // relu hidden (A of GEMM2)
  __shared__ float zS [MT][HH + 1];      // current state z
  __shared__ float ziS[MT][HH + 1];      // stage input; reused as k4 store
  __shared__ float k1S[MT][HH + 1];
  __shared__ float k2S[MT][HH + 1];
  __shared__ float k3S[MT][HH + 1];
  __shared__ float d0S[MT][CCH];         // diffs[i-1]      (dx/dt stage 0)
  __shared__ float dS [MT][CCH];         // diffs[i]        (dx/dt stages 1,3)
  __shared__ float wmS[MT][CCH];         // -1/3 d0 + 4/3 D (dx/dt stage 2)
  __shared__ float b1S[MLPW];
  __shared__ float b2S[ONN];

  const int tid  = threadIdx.x;
  const int lane = tid & 31;
  const int wave = tid >> 5;             // 0..7
  const int b0   = blockIdx.x * MT;

  // ---- stage weights into LDS, transposed to B-matrix [K][N] layout ----
  for (int i = tid; i < HH * MLPW; i += 256)   // W1[n][k], i = n*32+k
    W1t[i & (HH - 1)][i >> 5] = W1[i];
  for (int i = tid; i < MLPW * ONN; i += 256)  // W2[n][k], i = n*128+k
    W2t[i & (MLPW - 1)][i >> 7] = W2[i];
  if (tid < MLPW) b1S[tid] = b1[tid];
  b2S[tid] = b2[tid];

  // ---- z0 = X[:,0,:] @ Winit^T + binit  (tiny, scalar) ----
  for (int e = 0; e < 2; ++e) {
    int idx = tid + e * 256;
    int r = idx >> 5, h = idx & 31;
    const float* xr = X + (size_t)(b0 + r) * (LSEQ * CCH);
    float acc = binit[h];
#pragma unroll
    for (int c = 0; c < CCH; ++c) acc += xr[c] * Winit[h * CCH + c];
    zS[r][h] = acc;
  }

  // ---- diffs[0]; d0 at step 0 equals diffs[0] (=> stage-2 mix also = D) ----
  const int lr = tid >> 3, lc = tid & 7;
  float xprev = 0.0f;
  if (tid < MT * CCH) {
    const float* xr = X + (size_t)(b0 + lr) * (LSEQ * CCH) + lc;
    float d = xr[CCH] - xr[0];
    dS[lr][lc]  = d;
    d0S[lr][lc] = d;
    wmS[lr][lc] = d;
    xprev = xr[CCH];                     // x[l=1]
  }
  __syncthreads();

  for (int i = 0; i < LSEQ - 1; ++i) {
    // Prefetch x[i+2] for next step's diffs; latency hidden by 4 stages.
    float xnext = 0.0f;
    const bool haveNext = (i + 2) < LSEQ;
    if (tid < MT * CCH && haveNext)
      xnext = X[(size_t)(b0 + lr) * (LSEQ * CCH) + (size_t)(i + 2) * CCH + lc];

#pragma unroll 1
    for (int s = 0; s < 4; ++s) {
      if (s > 0) {                       // build stage input z_in
        for (int e = 0; e < 2; ++e) {
          int idx = tid + e * 256;
          int r = idx >> 5, h = idx & 31;
          float z = zS[r][h], zi;
          if (s == 1)      zi = z + k1S[r][h] * (1.0f / 3.0f);
          else if (s == 2) zi = z + k2S[r][h] - k1S[r][h] * (1.0f / 3.0f);
          else             zi = z + k1S[r][h] - k2S[r][h] + k3S[r][h];
          ziS[r][h] = zi;
        }
        __syncthreads();
      }
      const float (*zin)[HH + 1] = (s == 0) ? zS : ziS;

      // ---- GEMM1: h = relu(z_in @ W1^T + b1), 16x128, one 16-tile per wave.
      //      K=32 split into two independent 4-deep WMMA chains.
      {
        const int n0 = wave * 16;
        const float bias = b1S[n0 + (lane & 15)];
        v8f accL, accH;
#pragma unroll
        for (int j = 0; j < 8; ++j) { accL[j] = bias; accH[j] = 0.0f; }
#pragma unroll
        for (int k0 = 0; k0 < 16; k0 += 4) {
          v2f aL = frag_a(&zin[0][0], HH + 1, k0,      lane);
          v2f aH = frag_a(&zin[0][0], HH + 1, k0 + 16, lane);
          v2f bL = frag_b(&W1t[0][0], MLPW + 1, k0,      n0, lane);
          v2f bH = frag_b(&W1t[0][0], MLPW + 1, k0 + 16, n0, lane);
          accL = __builtin_amdgcn_wmma_f32_16x16x4_f32(false, aL, false, bL,
                                                       (short)0, accL, false, false);
          accH = __builtin_amdgcn_wmma_f32_16x16x4_f32(false, aH, false, bH,
                                                       (short)0, accH, false, false);
        }
        const int m0 = (lane >> 4) * 8;
        const int n  = n0 + (lane & 15);
#pragma unroll
        for (int j = 0; j < 8; ++j) hS[m0 + j][n] = fmaxf(accL[j] + accH[j], 0.0f);
      }
      __syncthreads();

      // ---- GEMM2 + tanh + dx/dt contraction -> k_s (16x32); 2 tiles/wave,
      //      shared A fragments, K=128 split -> 4 independent 16-deep chains.
      {
        float (*kt)[HH + 1] = (s == 0) ? k1S : (s == 1) ? k2S
                              : (s == 2) ? k3S : ziS;   // k4 reuses ziS
        const float* wrow = (s == 0) ? &d0S[0][0]
                          : (s == 2) ? &wmS[0][0] : &dS[0][0];
        const int n0a = wave * 32, n0b = n0a + 16;
        v8f a0L, a0H, a1L, a1H;
        const float ba0 = b2S[n0a + (lane & 15)];
        const float bb0 = b2S[n0b + (lane & 15)];
#pragma unroll
        for (int j = 0; j < 8; ++j) {
          a0L[j] = ba0; a0H[j] = 0.0f;
          a1L[j] = bb0; a1H[j] = 0.0f;
        }
#pragma unroll 4
        for (int k0 = 0; k0 < 64; k0 += 4) {
          v2f aL  = frag_a(&hS[0][0], MLPW + 1, k0,      lane);
          v2f aH  = frag_a(&hS[0][0], MLPW + 1, k0 + 64, lane);
          v2f bAL = frag_b(&W2t[0][0], ONN + 1, k0,      n0a, lane);
          v2f bAH = frag_b(&W2t[0][0], ONN + 1, k0 + 64, n0a, lane);
          v2f bBL = frag_b(&W2t[0][0], ONN + 1, k0,      n0b, lane);
          v2f bBH = frag_b(&W2t[0][0], ONN + 1, k0 + 64, n0b, lane);
          a0L = __builtin_amdgcn_wmma_f32_16x16x4_f32(false, aL, false, bAL,
                                                      (short)0, a0L, false, false);
          a0H = __builtin_amdgcn_wmma_f32_16x16x4_f32(false, aH, false, bAH,
                                                      (short)0, a0H, false, false);
          a1L = __builtin_amdgcn_wmma_f32_16x16x4_f32(false, aL, false, bBL,
                                                      (short)0, a1L, false, false);
          a1H = __builtin_amdgcn_wmma_f32_16x16x4_f32(false, aH, false, bBH,
                                                      (short)0, a1H, false, false);
        }
        const int m0 = (lane >> 4) * 8;
#pragma unroll
        for (int t = 0; t < 2; ++t) {
          const int col = ((t == 0) ? n0a : n0b) + (lane & 15);
          const int c = col & 7, h = col >> 3;   // out.reshape(B,H,C): n = h*8+c
          float v[8];
#pragma unroll
          for (int j = 0; j < 8; ++j) {
            float acc = (t == 0) ? (a0L[j] + a0H[j]) : (a1L[j] + a1H[j]);
            float wt  = wrow[(m0 + j) * CCH + c];
            v[j] = tanh_fast(acc) * wt;
          }
          // Reduce over c: butterfly across the 8-lane group (c dimension).
#pragma unroll
          for (int j = 0; j < 8; ++j) {
            v[j] += __shfl_xor(v[j], 1, 32);
            v[j] += __shfl_xor(v[j], 2, 32);
            v[j] += __shfl_xor(v[j], 4, 32);
          }
          if ((lane & 7) == 0) {
#pragma unroll
            for (int j = 0; j < 8; ++j) kt[m0 + j][h] = v[j];
          }
        }
      }
      __syncthreads();
    } // stages

    // ---- z += (k1 + 3(k2+k3) + k4)/8 ; rotate diffs + stage-2 mix ----
    for (int e = 0; e < 2; ++e) {
      int idx = tid + e * 256;
      int r = idx >> 5, h = idx & 31;
      float k1 = k1S[r][h], k2 = k2S[r][h], k3 = k3S[r][h], k4 = ziS[r][h];
      zS[r][h] += (k1 + 3.0f * (k2 + k3) + k4) * 0.125f;
    }
    if (tid < MT * CCH) {
      float dOld = dS[lr][lc];
      d0S[lr][lc] = dOld;
      if (haveNext) {
        float dNew = xnext - xprev;
        dS[lr][lc]  = dNew;
        wmS[lr][lc] = (4.0f / 3.0f) * dNew - (1.0f / 3.0f) * dOld;
        xprev = xnext;
      }
    }
    __syncthreads();
  }

  // ---- write z_T [B,32] ----
  for (int e = 0; e < 2; ++e) {
    int idx = tid + e * 256;
    int r = idx >> 5, h = idx & 31;
    out[(size_t)(b0 + r) * HH + h] = zS[r][h];
  }
}

extern "C" void kernel_launch(void* const* d_in, const int* in_sizes, int n_in,
                              void* d_out, int out_size, void* d_ws, size_t ws_size,
                              hipStream_t stream) {
  const float* X     = (const float*)d_in[0];
  const float* Winit = (const float*)d_in[1];
  const float* binit = (const float*)d_in[2];
  const float* W1    = (const float*)d_in[3];
  const float* b1    = (const float*)d_in[4];
  const float* W2    = (const float*)d_in[5];
  const float* b2    = (const float*)d_in[6];
  float* out = (float*)d_out;

  const int B    = in_sizes[0] / (LSEQ * CCH);   // 256
  const int grid = B / MT;                        // 16 independent batch tiles
  hipLaunchKernelGGL(cde_rk4_kernel, dim3(grid), dim3(256), 0, stream,
                     X, Winit, binit, W1, b1, W2, b2, out);
}